// CostDifference_61495341744708
// MI455X (gfx1250) — compile-verified
//
#include <hip/hip_runtime.h>
#include <stdint.h>

// Problem constants from the reference:
//   N=1, C=16, H=128, W=256, disparities D = -MIN_DISP = 128
//   out[d,c,h,w] = (h < d) ? left[c,h,w] - right[c, h+128-d, w] : 0
// Output layout (N,D,C,H,W) flat = ((d*C + c)*H + h)*W + w.
#define C_DIM  16
#define H_DIM  128
#define W_DIM  256
#define D_DISP 128

typedef float v4f __attribute__((ext_vector_type(4)));
// Exact pointee type expected by __builtin_amdgcn_global_load_async_to_lds_b128
// (per hipcc diagnostic: "__attribute__((__vector_size__(4 * sizeof(int)))) int __device__ *")
typedef int v4i_vs __attribute__((vector_size(16)));

#if defined(__HIP_DEVICE_COMPILE__) && __has_builtin(__builtin_amdgcn_global_load_async_to_lds_b128)
#define USE_ASYNC_LDS 1
#else
#define USE_ASYNC_LDS 0
#endif

__global__ __launch_bounds__(256) void cost_diff_kernel(
    const float* __restrict__ left,
    const float* __restrict__ right,
    float* __restrict__ out)
{
    const int tid = threadIdx.x;                 // 0..255 (8 wave32s)
    const int d   = blockIdx.z;                  // disparity 0..127
    const int c   = blockIdx.y;                  // channel 0..15
    const int h   = blockIdx.x * 4 + (tid >> 6); // 4 H-rows per block; uniform per wave
    const int w4  = tid & 63;                    // float4 column (64 per W-row)

    const size_t lidx = ((size_t)c * H_DIM + h) * W_DIM + (size_t)w4 * 4;
    const size_t oidx = (((size_t)d * C_DIM + c) * (size_t)H_DIM + h) * W_DIM + (size_t)w4 * 4;

    v4f res = {0.f, 0.f, 0.f, 0.f};

    if (h < d) {  // wave-uniform predicate: lanes in a wave share h, d
#if USE_ASYNC_LDS
        __shared__ v4f lds_left[256];
        typedef __attribute__((address_space(1))) v4i_vs* gv4p;  // global (__device__)
        typedef __attribute__((address_space(3))) v4i_vs* lv4p;  // LDS (__shared__)
        // Stage this thread's left float4 via the CDNA5 async DMA pipe
        // (global_load_async_to_lds_b128, tracked by ASYNCcnt).
        __builtin_amdgcn_global_load_async_to_lds_b128(
            (gv4p)(uintptr_t)(left + lidx),
            (lv4p)(uint32_t)(uintptr_t)(&lds_left[tid]),
            /*offset=*/0, /*cpol=*/0);
#endif
        // right[c, h+128-d, w]: valid row index in [128-d, 127]; cached (RT)
        // loads so the 2 MB right tensor stays L2-resident across its ~128x reuse.
        const size_t ridx = ((size_t)c * H_DIM + (h + H_DIM - d)) * W_DIM + (size_t)w4 * 4;
        const v4f r4 = *reinterpret_cast<const v4f*>(right + ridx);

#if USE_ASYNC_LDS
#if __has_builtin(__builtin_amdgcn_s_wait_asynccnt)
        __builtin_amdgcn_s_wait_asynccnt(0);
#else
        asm volatile("s_wait_asynccnt 0" ::: "memory");
#endif
        const v4f l4 = lds_left[tid];  // producer == consumer lane: no barrier needed
#else
        const v4f l4 = *reinterpret_cast<const v4f*>(left + lidx);
#endif
        res = l4 - r4;
    }

    // 256 MB streaming output: non-temporal B128 store so the write stream
    // doesn't thrash the L2 lines holding left/right.
    __builtin_nontemporal_store(res, reinterpret_cast<v4f*>(out + oidx));
}

extern "C" void kernel_launch(void* const* d_in, const int* in_sizes, int n_in,
                              void* d_out, int out_size, void* d_ws, size_t ws_size,
                              hipStream_t stream) {
    (void)in_sizes; (void)n_in; (void)out_size; (void)d_ws; (void)ws_size;
    const float* left  = (const float*)d_in[0];
    const float* right = (const float*)d_in[1];
    float* out = (float*)d_out;

    dim3 grid(H_DIM / 4, C_DIM, D_DISP);  // (32, 16, 128)
    dim3 block(256);
    cost_diff_kernel<<<grid, block, 0, stream>>>(left, right, out);
}